// MOE_DISTRIBUTE_CASCADE_GRAPH_Model_59270548684992
// MI455X (gfx1250) — compile-verified
//
#include <hip/hip_runtime.h>
#include <stdint.h>

// Problem constants (match reference: B=4096, K=8, H=4096, E=64)
constexpr int Bt = 4096;        // tokens
constexpr int Kk = 8;           // experts per token
constexpr int Hh = 4096;        // hidden dim
constexpr int Ee = 64;          // number of experts
constexpr int NSLOT = Bt * Kk;  // 32768 token-expert slots
constexpr int HALF = Hh / 2;    // 2048 (H processed in two passes -> 64KB LDS)

// ---------------------------------------------------------------------------
// Kernel 1: stable counting-sort ranks.
// inv[i] = #{j : e_j < e_i} + #{j < i : e_j == e_i}   (== argsort(argsort(e)))
// Single workgroup, 256 threads, each owns 128 consecutive slots.
// ---------------------------------------------------------------------------
__global__ __launch_bounds__(256) void moe_rank_kernel(
    const int* __restrict__ expert_ids,  // [NSLOT]
    int* __restrict__ inv)               // [NSLOT]
{
    __shared__ unsigned short hist[256][Ee];  // 32 KB
    __shared__ unsigned int   expertBase[Ee];

    const int t   = threadIdx.x;
    const int per = NSLOT / 256;  // 128
    const int base = t * per;

    #pragma unroll
    for (int e = 0; e < Ee; ++e) hist[t][e] = 0;
    __syncthreads();

    // Phase A: per-thread histogram (thread owns row t -> no races)
    for (int j = 0; j < per; ++j) {
        int e = expert_ids[base + j];
        hist[t][e] = (unsigned short)(hist[t][e] + 1);
    }
    __syncthreads();

    // Phase B: per-expert exclusive scan over threads (64 threads, one column each)
    if (t < Ee) {
        unsigned int run = 0;
        for (int tt = 0; tt < 256; ++tt) {
            unsigned int c = hist[tt][t];
            hist[tt][t] = (unsigned short)run;  // exclusive base for (thread tt, expert t)
            run += c;
        }
        expertBase[t] = run;  // total count for expert t (scanned below)
    }
    __syncthreads();

    // Phase C: exclusive scan over experts (tiny, serial on thread 0)
    if (t == 0) {
        unsigned int run = 0;
        for (int e = 0; e < Ee; ++e) {
            unsigned int c = expertBase[e];
            expertBase[e] = run;
            run += c;
        }
    }
    __syncthreads();

    // Phase D: deterministic in-order rank assignment (stable)
    for (int j = 0; j < per; ++j) {
        int e = expert_ids[base + j];
        unsigned int r = expertBase[e] + (unsigned int)hist[t][e];
        hist[t][e] = (unsigned short)(hist[t][e] + 1);
        inv[base + j] = (int)r;
    }
}

// ---------------------------------------------------------------------------
// Kernel 2: combine. One block per token.
//   out[b,h] = sum_k scales[b,k] * expand_x[inv[b*K+k], h]   (written twice)
// Expert rows are staged through LDS with gfx1250 async global->LDS loads
// (global_load_async_to_lds_b128, tracked by ASYNCcnt). Each lane reads back
// exactly the bytes it fetched, so no workgroup barriers are required.
// ---------------------------------------------------------------------------
__global__ __launch_bounds__(256) void moe_combine_kernel(
    const float* __restrict__ expand_x,  // [NSLOT, H]
    const float* __restrict__ scales,    // [B, K]
    const int*   __restrict__ inv,       // [NSLOT]
    float* __restrict__ out)             // [2, B, H]
{
    __shared__ float buf[Kk][HALF];  // 8 rows x 2048 floats = 64 KB

    const int b = blockIdx.x;
    const int t = threadIdx.x;

    // Row base addresses + routing scales (uniform across the block)
    uint64_t rowaddr[Kk];
    float    s[Kk];
    #pragma unroll
    for (int k = 0; k < Kk; ++k) {
        const int r   = inv[b * Kk + k];
        rowaddr[k]    = (uint64_t)(uintptr_t)(expand_x + (size_t)r * Hh);
        s[k]          = scales[b * Kk + k];
    }

    #pragma unroll
    for (int pass = 0; pass < 2; ++pass) {
        const int hbase = pass * HALF;

        if (pass != 0) {
            // Ensure this wave's LDS reads from the previous pass completed
            // before the async engine overwrites the buffers.
            asm volatile("s_wait_dscnt 0" ::: "memory");
        }

        // Issue 8 rows x 2 chunks of 16B per lane: 16 async loads in flight.
        #pragma unroll
        for (int k = 0; k < Kk; ++k) {
            const uint32_t ldsbase = (uint32_t)(uintptr_t)(&buf[k][0]);
            const uint64_t gbase   = rowaddr[k] + (uint64_t)hbase * 4u;
            #pragma unroll
            for (int i = 0; i < 2; ++i) {
                const uint32_t off = (uint32_t)((i * 256 + t) * 16);
                const uint32_t lds = ldsbase + off;
                const uint64_t ga  = gbase + off;
                asm volatile("global_load_async_to_lds_b128 %0, %1, off"
                             :: "v"(lds), "v"(ga)
                             : "memory");
            }
        }

        // Wait for all async transfers of this pass to land in LDS.
        asm volatile("s_wait_asynccnt 0" ::: "memory");

        // Weighted accumulation over K from LDS.
        float4 acc[2];
        #pragma unroll
        for (int i = 0; i < 2; ++i) acc[i] = make_float4(0.f, 0.f, 0.f, 0.f);

        #pragma unroll
        for (int k = 0; k < Kk; ++k) {
            const float sk = s[k];
            #pragma unroll
            for (int i = 0; i < 2; ++i) {
                const float4 v = *(const float4*)&buf[k][(i * 256 + t) * 4];
                acc[i].x += sk * v.x;
                acc[i].y += sk * v.y;
                acc[i].z += sk * v.z;
                acc[i].w += sk * v.w;
            }
        }

        // Store to both tuple outputs (concatenated flat in d_out).
        #pragma unroll
        for (int i = 0; i < 2; ++i) {
            const int h = hbase + (i * 256 + t) * 4;
            *(float4*)&out[(size_t)b * Hh + h]        = acc[i];
            *(float4*)&out[(size_t)(Bt + b) * Hh + h] = acc[i];
        }
    }
}

// ---------------------------------------------------------------------------
// Host launcher.
// Inputs (setup_inputs order): x[B,H] f32 (unused after golden override),
// expert_ids[B,K] i32, expert_scales[B,K] f32, golden_expand_x[B*K,H] f32,
// moe_expert_num (scalar i32).
// d_out: 2*B*H f32 (tuple of two identical outputs).
// ---------------------------------------------------------------------------
extern "C" void kernel_launch(void* const* d_in, const int* in_sizes, int n_in,
                              void* d_out, int out_size, void* d_ws, size_t ws_size,
                              hipStream_t stream) {
    (void)in_sizes; (void)n_in; (void)out_size; (void)ws_size;

    const int*   expert_ids = (const int*)d_in[1];
    const float* scales     = (const float*)d_in[2];
    const float* expand_x   = (const float*)d_in[3];

    int* inv = (int*)d_ws;  // 32768 ints = 128 KB of scratch

    moe_rank_kernel<<<1, 256, 0, stream>>>(expert_ids, inv);
    moe_combine_kernel<<<Bt, 256, 0, stream>>>(expand_x, scales, inv, (float*)d_out);
}